// FRCLoss_59811714564599
// MI455X (gfx1250) — compile-verified
//
#include <hip/hip_runtime.h>
#include <hip/hip_bf16.h>
#include <math.h>

// ---------------------------------------------------------------------------
// FRC loss via DFT-as-matmul on CDNA5 WMMA (V_WMMA_F32_16X16X4_F32), with
// Tensor-Data-Mover staging of the shared B-operand strips into LDS.
//   fft2(X) = M @ X @ M,  M[a,b] = exp(-2*pi*i*a*b/512) = cos(t) - i*sin(t)
//   Stage1 (row pass):  G = X @ M   -> Gr = X@C, Gi = -(X@S)
//   Stage2 (col pass):  F = M @ G   -> Fr = C@Gr + S@Gi, Fi = C@Gi - S@Gr
// Block layout: one 16-column strip x 128 rows; 8 waves x one 16x16 tile each.
// The 512x16 B strip is DMA'd once per block into LDS (TDM, TENSORcnt), then
// every K-step reads B-fragments from LDS (ds_load) instead of strided VMEM.
// Stage1 uses the symmetry of M to fetch the strip as contiguous rows.
// ---------------------------------------------------------------------------

typedef __attribute__((ext_vector_type(2))) float v2f;
typedef __attribute__((ext_vector_type(8))) float v8f;
typedef unsigned int u32x4 __attribute__((ext_vector_type(4)));
typedef int i32x4 __attribute__((ext_vector_type(4)));
typedef int i32x8 __attribute__((ext_vector_type(8)));

#define N512   512
#define IMG    (N512 * N512)          // floats per plane
#define NIMG   48                     // B*C = 16*3
#define NSEG   254                    // R_MAX - 2

// Stage1 LDS strip: 16 rows x (512 + 2-dword pad per 256 dwords) = 16 x 516
#define S1_ROW   516
#define S1_STRIP (16 * S1_ROW)        // floats per matrix strip
// Stage2 LDS strip: [512][16] natural layout
#define S2_STRIP (N512 * 16)

#if __has_builtin(__builtin_amdgcn_tensor_load_to_lds)
#  define FRC_HAS_TDM 1
#  if __has_include(<hip/amd_detail/amd_gfx1250_TDM.h>)
#    define FRC_TDM_6ARG 1
#  endif
#endif

// ---------------------------------------------------------------------------
// TDM 2D tile load: tensor is a 512x512 f32 row-major matrix (stride 512),
// tile is tile1 rows x tile0 contiguous elements starting at gptr.
// flags = group1 dword0 (data_size / pad config).  CDNA5 ISA.
// ---------------------------------------------------------------------------
#if defined(FRC_HAS_TDM)
__device__ __forceinline__ void tdm_load_2d(unsigned lds_off_bytes,
                                            const float* gptr,
                                            unsigned flags,
                                            unsigned tile0, unsigned tile1) {
    unsigned long long ga = (unsigned long long)(uintptr_t)gptr;
    u32x4 g0;
    g0[0] = 1u;                                          // count=1, user D#
    g0[1] = lds_off_bytes;                               // lds_addr
    g0[2] = (unsigned)(ga & 0xFFFFFFFFu);                // global_addr lo
    g0[3] = (unsigned)((ga >> 32) & 0x01FFFFFFu) | (2u << 30);  // addr hi | type=2
    i32x8 g1;
    g1[0] = (int)flags;              // wg_mask=0 | data_size=4B | pad cfg
    g1[1] = (int)(512u << 16);       // [31:16] = tensor_dim0 lo16 (=512)
    g1[2] = (int)(512u << 16);       // dim0 hi=0 | [31:16] tensor_dim1 lo16
    g1[3] = (int)(tile0 << 16);      // dim1 hi=0 | [31:16] tile_dim0
    g1[4] = (int)(tile1 & 0xFFFFu);  // [15:0] tile_dim1 | tile_dim2=0
    g1[5] = 512;                     // tensor_dim0_stride lo32
    g1[6] = (int)(512u << 16);       // stride0 hi=0 | tensor_dim1_stride lo16
    g1[7] = 0;
    i32x4 z4 = {0, 0, 0, 0};
#if defined(FRC_TDM_6ARG)
    i32x8 z8 = {0, 0, 0, 0, 0, 0, 0, 0};
    __builtin_amdgcn_tensor_load_to_lds(g0, g1, z4, z4, z8, 0);
#else
    __builtin_amdgcn_tensor_load_to_lds(g0, g1, z4, z4, 0);
#endif
}
#endif

// group1 dword0 values: data_size=2(4B)<<16 [+ pad_en<<20 | interval7<<22 | amount1<<25]
#define TDM_FLAGS_PLAIN  0x00020000u
#define TDM_FLAGS_PAD    (0x00020000u | (1u << 20) | (7u << 22) | (1u << 25))

// ---------------------------------------------------------------------------
// Build cos/sin DFT matrices: C[a,b]=cos(2pi ab/512), S[a,b]=sin(2pi ab/512)
// (phase reduced mod 512 in integers for f32 accuracy).
// ---------------------------------------------------------------------------
__global__ void frc_build_dft(float* __restrict__ C, float* __restrict__ S) {
    int idx = blockIdx.x * blockDim.x + threadIdx.x;
    if (idx >= IMG) return;
    int a = idx >> 9;
    int b = idx & 511;
    int p = (a * b) & 511;
    float ang = (float)p * (6.2831853071795864769f / 512.0f);
    C[idx] = cosf(ang);
    S[idx] = sinf(ang);
}

// ---------------------------------------------------------------------------
// Stage 1: G = X @ M per image.  Block: img / colstrip / rowgroup.
// LDS holds rows col0..col0+15 of Mc and Ms (symmetric => B columns).
// Padded row stride 516 keeps transposed b64 reads aligned & bank-spread.
// ---------------------------------------------------------------------------
__global__ void __launch_bounds__(256)
frc_stage1(const float* __restrict__ X,      // [NIMG][512][512]
           const float* __restrict__ Mc,
           const float* __restrict__ Ms,
           float* __restrict__ G)            // [NIMG][2][512][512]
{
    extern __shared__ float smem[];
    float* sc = smem;                 // [16][516]
    float* ss = smem + S1_STRIP;

    const int lane = threadIdx.x & 31;
    const int wave = threadIdx.x >> 5;
    const int img  = blockIdx.x >> 7;            // /128
    const int rem  = blockIdx.x & 127;
    const int col0 = (rem & 31) << 4;
    const int row0 = ((rem >> 5) << 7) + (wave << 4);

#if defined(FRC_HAS_TDM)
    if (wave == 0) {
        // Row strips (contiguous) thanks to M symmetry; padded into LDS.
        tdm_load_2d(0u,                      Mc + (size_t)col0 * N512,
                    TDM_FLAGS_PAD, 512u, 16u);
        tdm_load_2d((unsigned)(S1_STRIP * 4), Ms + (size_t)col0 * N512,
                    TDM_FLAGS_PAD, 512u, 16u);
        __builtin_amdgcn_s_wait_tensorcnt(0);
    }
    __syncthreads();
#else
    for (int i = threadIdx.x; i < 16 * N512; i += 256) {
        int r = i >> 9, c = i & 511;
        int d = r * S1_ROW + c + ((c >> 8) << 1);
        sc[d] = Mc[(size_t)(col0 + r) * N512 + c];
        ss[d] = Ms[(size_t)(col0 + r) * N512 + c];
    }
    __syncthreads();
#endif

    const float* x = X + (size_t)img * IMG;
    const int m  = lane & 15;
    const int kk = (lane >> 4) << 1;

    v8f accR = {0.f,0.f,0.f,0.f,0.f,0.f,0.f,0.f};
    v8f accI = {0.f,0.f,0.f,0.f,0.f,0.f,0.f,0.f};

    const int srow = m * S1_ROW;      // this lane's strip row (output column)
    for (int k = 0; k < N512; k += 4) {
        v2f a = *(const v2f*)(x + (size_t)(row0 + m) * N512 + (k + kk));
        int e = k + kk;
        int d = srow + e + ((e >> 8) << 1);    // padded, even -> 8B aligned
        v2f bc = *(const v2f*)(sc + d);        // B[k..k+1][col0+m] via symmetry
        v2f bs = *(const v2f*)(ss + d);
        accR = __builtin_amdgcn_wmma_f32_16x16x4_f32(false, a, false, bc,
                                                     (short)0, accR, false, false);
        accI = __builtin_amdgcn_wmma_f32_16x16x4_f32(false, a, false, bs,
                                                     (short)0, accI, false, false);
    }

    float* gr = G + ((size_t)img * 2 + 0) * IMG;
    float* gi = G + ((size_t)img * 2 + 1) * IMG;
    const int roff = (lane >> 4) << 3;
    const int cc   = col0 + (lane & 15);
#pragma unroll
    for (int g = 0; g < 8; ++g) {
        size_t off = (size_t)(row0 + roff + g) * N512 + cc;
        gr[off] =  accR[g];
        gi[off] = -accI[g];   // Gi = -(X@S)
    }
}

// ---------------------------------------------------------------------------
// Stage 2: F = M @ G per image.  LDS holds the [512][16] column strips of
// Gr and Gi (natural layout, bank-conflict-free).  A rows are contiguous.
// Fr = C@Gr + S@Gi, Fi = C@Gi - S@Gr (f32 WMMA has no A/B negate).
// ---------------------------------------------------------------------------
__global__ void __launch_bounds__(256)
frc_stage2(const float* __restrict__ G,      // [NIMG][2][512][512]
           const float* __restrict__ Mc,
           const float* __restrict__ Ms,
           float* __restrict__ F)            // [NIMG][2][512][512]
{
    extern __shared__ float smem[];
    float* sgr = smem;                // [512][16]
    float* sgi = smem + S2_STRIP;

    const int lane = threadIdx.x & 31;
    const int wave = threadIdx.x >> 5;
    const int img  = blockIdx.x >> 7;
    const int rem  = blockIdx.x & 127;
    const int col0 = (rem & 31) << 4;
    const int row0 = ((rem >> 5) << 7) + (wave << 4);

    const float* gr = G + ((size_t)img * 2 + 0) * IMG;
    const float* gi = G + ((size_t)img * 2 + 1) * IMG;

#if defined(FRC_HAS_TDM)
    if (wave == 0) {
        tdm_load_2d(0u,                      gr + col0, TDM_FLAGS_PLAIN, 16u, 512u);
        tdm_load_2d((unsigned)(S2_STRIP * 4), gi + col0, TDM_FLAGS_PLAIN, 16u, 512u);
        __builtin_amdgcn_s_wait_tensorcnt(0);
    }
    __syncthreads();
#else
    for (int i = threadIdx.x; i < S2_STRIP; i += 256) {
        int r = i >> 4, c = i & 15;
        sgr[i] = gr[(size_t)r * N512 + col0 + c];
        sgi[i] = gi[(size_t)r * N512 + col0 + c];
    }
    __syncthreads();
#endif

    const int m  = lane & 15;
    const int kk = (lane >> 4) << 1;

    v8f aCR = {0.f,0.f,0.f,0.f,0.f,0.f,0.f,0.f};
    v8f aSI = aCR, aCI = aCR, aSR = aCR;

    for (int k = 0; k < N512; k += 4) {
        v2f ac = *(const v2f*)(Mc + (size_t)(row0 + m) * N512 + (k + kk));
        v2f as = *(const v2f*)(Ms + (size_t)(row0 + m) * N512 + (k + kk));
        v2f br, bi;
        br.x = sgr[(k + kk)     * 16 + m];
        br.y = sgr[(k + kk + 1) * 16 + m];
        bi.x = sgi[(k + kk)     * 16 + m];
        bi.y = sgi[(k + kk + 1) * 16 + m];
        aCR = __builtin_amdgcn_wmma_f32_16x16x4_f32(false, ac, false, br,
                                                    (short)0, aCR, false, false);
        aSI = __builtin_amdgcn_wmma_f32_16x16x4_f32(false, as, false, bi,
                                                    (short)0, aSI, false, false);
        aCI = __builtin_amdgcn_wmma_f32_16x16x4_f32(false, ac, false, bi,
                                                    (short)0, aCI, false, false);
        aSR = __builtin_amdgcn_wmma_f32_16x16x4_f32(false, as, false, br,
                                                    (short)0, aSR, false, false);
    }

    float* fr = F + ((size_t)img * 2 + 0) * IMG;
    float* fi = F + ((size_t)img * 2 + 1) * IMG;
    const int roff = (lane >> 4) << 3;
    const int cc   = col0 + (lane & 15);
#pragma unroll
    for (int g = 0; g < 8; ++g) {
        size_t off = (size_t)(row0 + roff + g) * N512 + cc;
        fr[off] = aCR[g] + aSI[g];
        fi[off] = aCI[g] - aSR[g];
    }
}

// ---------------------------------------------------------------------------
// Segment gather: one thread per (perimeter point, image).  Duplicated
// Bresenham points are counted exactly like jax segment_sum.
// accum[seg][img] = {cross_re, cross_im, e1, e2}
// ---------------------------------------------------------------------------
__global__ void frc_zero(float* __restrict__ acc, int n) {
    int i = blockIdx.x * blockDim.x + threadIdx.x;
    if (i < n) acc[i] = 0.f;
}

__global__ void frc_gather(const float* __restrict__ Fin,
                           const float* __restrict__ Ftg,
                           const int* __restrict__ rows,
                           const int* __restrict__ cols,
                           const int* __restrict__ segs,
                           int P,
                           float* __restrict__ accum)
{
    int idx = blockIdx.x * blockDim.x + threadIdx.x;
    int total = P * NIMG;
    if (idx >= total) return;
    int p  = idx / NIMG;
    int bc = idx - p * NIMG;
    int r = rows[p], c = cols[p], s = segs[p];

    size_t off = (size_t)bc * 2 * IMG + (size_t)r * N512 + c;
    float ar = Fin[off], ai = Fin[off + IMG];
    float br = Ftg[off], bi = Ftg[off + IMG];

    float cr = ar * br + ai * bi;     // Re(p1 * conj(p2))
    float ci = ai * br - ar * bi;     // Im(p1 * conj(p2))
    float e1 = ar * ar + ai * ai;
    float e2 = br * br + bi * bi;

    float* acc = accum + ((size_t)s * NIMG + bc) * 4;
    atomicAdd(acc + 0, cr);
    atomicAdd(acc + 1, ci);
    atomicAdd(acc + 2, e1);
    atomicAdd(acc + 3, e2);
}

// ---------------------------------------------------------------------------
// Finalize: curve[0]=1, curve[1+s]=|cross|/sqrt(e1*e2); out = curve@W^T + b
// ---------------------------------------------------------------------------
__global__ void frc_finalize(const float* __restrict__ accum,
                             const float* __restrict__ weight,  // [1][255]
                             const float* __restrict__ bias,    // [1]
                             float* __restrict__ out)           // [48]
{
    int bc = blockIdx.x * blockDim.x + threadIdx.x;
    if (bc >= NIMG) return;
    float s = weight[0] * 1.0f;
    for (int seg = 0; seg < NSEG; ++seg) {
        const float* a = accum + ((size_t)seg * NIMG + bc) * 4;
        float mag = sqrtf(a[0] * a[0] + a[1] * a[1]);
        s += weight[seg + 1] * (mag * rsqrtf(a[2] * a[3]));
    }
    out[bc] = s + bias[0];
}

// ---------------------------------------------------------------------------
// Launch. Inputs: 0=input, 1=target, 2=rows, 3=cols, 4=segs, 5=weight, 6=bias
// ws layout: Mc | Ms | G(reused) | F_in | F_tgt | accum  (~304 MB)
// ---------------------------------------------------------------------------
extern "C" void kernel_launch(void* const* d_in, const int* in_sizes, int n_in,
                              void* d_out, int out_size, void* d_ws, size_t ws_size,
                              hipStream_t stream) {
    const float* inp    = (const float*)d_in[0];
    const float* tgt    = (const float*)d_in[1];
    const int*   rows   = (const int*)d_in[2];
    const int*   cols   = (const int*)d_in[3];
    const int*   segs   = (const int*)d_in[4];
    const float* weight = (const float*)d_in[5];
    const float* bias   = (const float*)d_in[6];
    float*       out    = (float*)d_out;
    const int P = in_sizes[2];

    char* ws = (char*)d_ws;
    const size_t planeB = (size_t)IMG * sizeof(float);       // 1 MB
    float* Mc    = (float*)(ws);
    float* Ms    = (float*)(ws + planeB);
    float* G     = (float*)(ws + 2 * planeB);                // 96 MB, reused
    float* Fin   = (float*)(ws + 2 * planeB + (size_t)NIMG * 2 * planeB);
    float* Ftg   = (float*)(ws + 2 * planeB + (size_t)NIMG * 4 * planeB);
    float* accum = (float*)(ws + 2 * planeB + (size_t)NIMG * 6 * planeB);

    // 1) DFT matrices
    frc_build_dft<<<(IMG + 255) / 256, 256, 0, stream>>>(Mc, Ms);

    // 2/3) row pass -> G, col pass -> F (input then target, G reused)
    const int tileBlocks = NIMG * 128;                       // img x colstrip x rowgroup
    const size_t lds1 = (size_t)2 * S1_STRIP * sizeof(float);   // 66048 B
    const size_t lds2 = (size_t)2 * S2_STRIP * sizeof(float);   // 65536 B
    frc_stage1<<<tileBlocks, 256, lds1, stream>>>(inp, Mc, Ms, G);
    frc_stage2<<<tileBlocks, 256, lds2, stream>>>(G, Mc, Ms, Fin);
    frc_stage1<<<tileBlocks, 256, lds1, stream>>>(tgt, Mc, Ms, G);
    frc_stage2<<<tileBlocks, 256, lds2, stream>>>(G, Mc, Ms, Ftg);

    // 4) segment reduction
    const int nacc = NSEG * NIMG * 4;
    frc_zero<<<(nacc + 255) / 256, 256, 0, stream>>>(accum, nacc);
    long long totalG = (long long)P * NIMG;
    frc_gather<<<(int)((totalG + 255) / 256), 256, 0, stream>>>(
        Fin, Ftg, rows, cols, segs, P, accum);

    // 5) FRC curve -> linear layer
    frc_finalize<<<1, 64, 0, stream>>>(accum, weight, bias, out);
}